// GCNMLP_32057635897406
// MI455X (gfx1250) — compile-verified
//
#include <hip/hip_runtime.h>
#include <math.h>

typedef __attribute__((ext_vector_type(2))) float v2f;
typedef __attribute__((ext_vector_type(8))) float v8f;

#define IN_CH 128
#define HID 64
#define NODES_PER_GRAPH 32

// ---------------------------------------------------------------- degree pass
__global__ void k_deg_init(float* __restrict__ deg, int n) {
  int i = blockIdx.x * blockDim.x + threadIdx.x;
  if (i < n) deg[i] = 1.0f;                       // self-loop weight
}

__global__ void k_deg_scatter(const int* __restrict__ col,
                              const float* __restrict__ w,
                              float* __restrict__ deg, int E) {
  int e = blockIdx.x * blockDim.x + threadIdx.x;
  if (e < E) atomicAdd(&deg[col[e]], w[e]);
}

__global__ void k_dis(float* __restrict__ deg, int n) {
  int i = blockIdx.x * blockDim.x + threadIdx.x;
  if (i < n) {
    float d = deg[i];
    deg[i] = (d > 0.0f) ? rsqrtf(d) : 0.0f;       // overwrite deg with D^-1/2
  }
}

// ------------------------------------------------- WMMA f32 GEMM: Out[M,64] =
// X[M,K] @ W[K,64] (+bias, +relu).  One wave per 16x16 output tile.
// A fragment (16x4 f32): lanes 0-15 -> K=k0,k0+1 ; lanes 16-31 -> K=k0+2,k0+3
// B fragment (4x16 f32): mirrored; C/D: 8 VGPRs, row = r + 8*(lane>>4).
__global__ __launch_bounds__(256) void k_gemm_wmma(
    const float* __restrict__ X, const float* __restrict__ W,
    const float* __restrict__ bias, float* __restrict__ Out,
    int M, int K, int doRelu) {
  const int lane = threadIdx.x & 31;
  const int wave = (blockIdx.x * blockDim.x + threadIdx.x) >> 5;
  const int tn   = wave & 3;            // 4 col tiles (N = 64)
  const int tm   = wave >> 2;           // row tile
  const int m16  = lane & 15;
  const int hi   = lane >> 4;           // 0 | 1
  const int mrow = tm * 16 + m16;
  const int ncol = tn * 16 + m16;
  const int kh   = hi * 2;

  const float* xrow = X + (size_t)mrow * K;
  v8f acc = {};
  for (int k0 = 0; k0 < K; k0 += 4) {
    v2f a, b;
    a.x = xrow[k0 + kh];                              // contiguous b64 per lane
    a.y = xrow[k0 + kh + 1];
    b.x = W[(size_t)(k0 + kh) * HID + ncol];
    b.y = W[(size_t)(k0 + kh + 1) * HID + ncol];
    acc = __builtin_amdgcn_wmma_f32_16x16x4_f32(false, a, false, b,
                                                (short)0, acc, false, false);
  }

  const int rbase = tm * 16 + 8 * hi;
  const float b0 = bias ? bias[ncol] : 0.0f;
#pragma unroll
  for (int r = 0; r < 8; ++r) {
    float v = acc[r] + b0;
    if (doRelu) v = fmaxf(v, 0.0f);
    Out[(size_t)(rbase + r) * HID + ncol] = v;
  }
}

// ------------------------------------- out = dis[n]^2 * h[n] + bg (self loop)
__global__ void k_selfloop_init(const float* __restrict__ dis,
                                const float* __restrict__ h,
                                const float* __restrict__ bg,
                                float* __restrict__ out, int Nn) {
  int t = blockIdx.x * blockDim.x + threadIdx.x;
  int n = t >> 4;                                   // 16 lanes per node
  int q = t & 15;                                   // 4 channels each
  if (n >= Nn) return;
  float s = dis[n]; s = s * s;
  const float4 hv = *(const float4*)(h + (size_t)n * HID + q * 4);
  const float4 bv = *(const float4*)(bg + q * 4);
  float4 o;
  o.x = s * hv.x + bv.x; o.y = s * hv.y + bv.y;
  o.z = s * hv.z + bv.z; o.w = s * hv.w + bv.w;
  *(float4*)(out + (size_t)n * HID + q * 4) = o;
}

// ------------------------------------------------- edge gather-scale-scatter
__global__ void k_edge_scatter(const int* __restrict__ row,
                               const int* __restrict__ col,
                               const float* __restrict__ w,
                               const float* __restrict__ dis,
                               const float* __restrict__ h,
                               float* __restrict__ out, int E) {
  int t = blockIdx.x * blockDim.x + threadIdx.x;
  int e = t >> 4;                                   // 16 lanes per edge
  int q = t & 15;
  if (e >= E) return;
  int r = row[e], c = col[e];
  float norm = dis[r] * w[e] * dis[c];
  const float4 hv = *(const float4*)(h + (size_t)r * HID + q * 4);
  float* op = out + (size_t)c * HID + q * 4;
  atomicAdd(op + 0, norm * hv.x);
  atomicAdd(op + 1, norm * hv.y);
  atomicAdd(op + 2, norm * hv.z);
  atomicAdd(op + 3, norm * hv.w);
}

// ------------------------------------------------------------- per-graph head
__global__ __launch_bounds__(32) void k_head(
    const float* __restrict__ z1, const float* __restrict__ W2,
    const float* __restrict__ b2, const float* __restrict__ W3,
    const float* __restrict__ b3, float* __restrict__ out) {
  __shared__ float s1[64];
  __shared__ float s2[32];
  __shared__ float s3[10];
  const int g = blockIdx.x;
  const int t = threadIdx.x;                        // 0..31, one wave32
  const float* z = z1 + (size_t)g * 64;
  s1[t] = z[t]; s1[t + 32] = z[t + 32];
  __syncthreads();
  float acc = b2[t];
#pragma unroll
  for (int k = 0; k < 64; ++k) acc += s1[k] * W2[k * 32 + t];
  s2[t] = fmaxf(acc, 0.0f);
  __syncthreads();
  if (t < 10) {
    float a3 = b3[t];
#pragma unroll
    for (int j = 0; j < 32; ++j) a3 += s2[j] * W3[j * 10 + t];
    s3[t] = a3;
  }
  __syncthreads();
  if (t < 10) {
    float mx = s3[0];
    for (int c = 1; c < 10; ++c) mx = fmaxf(mx, s3[c]);
    float sum = 0.0f;
    for (int c = 0; c < 10; ++c) sum += expf(s3[c] - mx);
    out[(size_t)g * 10 + t] = expf(s3[t] - mx) / sum;
  }
}

// ---------------------------------------------------------------------- entry
extern "C" void kernel_launch(void* const* d_in, const int* in_sizes, int n_in,
                              void* d_out, int out_size, void* d_ws, size_t ws_size,
                              hipStream_t stream) {
  const float* x  = (const float*)d_in[0];
  const int*   ei = (const int*)  d_in[1];
  const float* ew = (const float*)d_in[2];
  const float* Wg = (const float*)d_in[3];
  const float* bg = (const float*)d_in[4];
  const float* W1 = (const float*)d_in[5];
  const float* b1 = (const float*)d_in[6];
  const float* W2 = (const float*)d_in[7];
  const float* b2 = (const float*)d_in[8];
  const float* W3 = (const float*)d_in[9];
  const float* b3 = (const float*)d_in[10];
  float* out = (float*)d_out;

  const int Nn = in_sizes[0] / IN_CH;          // 262144 nodes
  const int E  = in_sizes[2];                  // 2097152 edges
  const int G  = Nn / NODES_PER_GRAPH;         // 8192 graphs
  const int* row = ei;
  const int* col = ei + E;

  // workspace layout: dis[Nn] | h[Nn*64] | outh[Nn*64] | z1[G*64]
  char* ws = (char*)d_ws;
  float* dis  = (float*)ws;
  float* h    = (float*)(ws + (size_t)Nn * 4);
  float* outh = (float*)(ws + (size_t)Nn * 4 + (size_t)Nn * HID * 4);
  float* z1   = (float*)(ws + (size_t)Nn * 4 + 2 * (size_t)Nn * HID * 4);

  // 1) symmetric-norm degrees (self loop weight 1), then D^-1/2 in place
  k_deg_init   <<<(Nn + 255) / 256, 256, 0, stream>>>(dis, Nn);
  k_deg_scatter<<<(E  + 255) / 256, 256, 0, stream>>>(col, ew, dis, E);
  k_dis        <<<(Nn + 255) / 256, 256, 0, stream>>>(dis, Nn);

  // 2) h = x @ Wg   (WMMA f32 16x16x4; 65536 waves, 8 waves/block)
  {
    int waves = (Nn / 16) * (HID / 16);
    k_gemm_wmma<<<waves / 8, 256, 0, stream>>>(x, Wg, nullptr, h, Nn, IN_CH, 0);
  }

  // 3) out = dis^2 * h + bg, then scatter-add edge contributions
  k_selfloop_init<<<(Nn * 16 + 255) / 256, 256, 0, stream>>>(dis, h, bg, outh, Nn);
  k_edge_scatter <<<(int)(((size_t)E * 16 + 255) / 256), 256, 0, stream>>>(
      row, col, ew, dis, h, outh, E);

  // 4) z1 = relu(Hflat[G,2048] @ W1 + b1)  (same WMMA kernel, K = 2048)
  {
    int waves = (G / 16) * (HID / 16);
    k_gemm_wmma<<<waves / 8, 256, 0, stream>>>(outh, W1, b1, z1, G,
                                               NODES_PER_GRAPH * HID, 1);
  }

  // 5) per-graph MLP tail + softmax
  k_head<<<G, 32, 0, stream>>>(z1, W2, b2, W3, b3, out);
}